// CNNWithHistogramPooling_89498528514148
// MI455X (gfx1250) — compile-verified
//
#include <hip/hip_runtime.h>
#include <hip/hip_bf16.h>

#define BINS 64
#define C_OUT 64
#define B_N 64
#define IW 256
#define OW 254
#define NPIX (OW * OW)

typedef __attribute__((ext_vector_type(2))) float v2f;
typedef __attribute__((ext_vector_type(8))) float v8f;

// ---------------- Pass 1: fused conv+relu, per-(b,c) min/max ----------------
__global__ void conv_minmax_kernel(const float* __restrict__ x,
                                   const float* __restrict__ w,
                                   const float* __restrict__ bias,
                                   float* __restrict__ mn,
                                   float* __restrict__ mx) {
    const int bc = blockIdx.x;           // bc = b*64 + c
    const int b  = bc >> 6;
    const int c  = bc & 63;
    const int tid = threadIdx.x;

    const float* wc = w + c * 9;
    const float w0 = wc[0], w1 = wc[1], w2 = wc[2];
    const float w3 = wc[3], w4 = wc[4], w5 = wc[5];
    const float w6 = wc[6], w7 = wc[7], w8 = wc[8];
    const float bb = bias[c];
    const float* xb = x + (size_t)b * IW * IW;

    float lmin =  3.4e38f;
    float lmax = -3.4e38f;

    for (int p = tid; p < NPIX; p += blockDim.x) {
        const int i = p / OW;
        const int j = p - i * OW;
        const float* r0 = xb + i * IW + j;
        float v = bb;
        v = fmaf(w0, r0[0],       v);
        v = fmaf(w1, r0[1],       v);
        v = fmaf(w2, r0[2],       v);
        v = fmaf(w3, r0[IW + 0],  v);
        v = fmaf(w4, r0[IW + 1],  v);
        v = fmaf(w5, r0[IW + 2],  v);
        v = fmaf(w6, r0[2*IW + 0], v);
        v = fmaf(w7, r0[2*IW + 1], v);
        v = fmaf(w8, r0[2*IW + 2], v);
        v = fmaxf(v, 0.0f);
        lmin = fminf(lmin, v);
        lmax = fmaxf(lmax, v);
    }

    __shared__ float smin[256];
    __shared__ float smax[256];
    smin[tid] = lmin;
    smax[tid] = lmax;
    __syncthreads();
    for (int s = 128; s > 0; s >>= 1) {
        if (tid < s) {
            smin[tid] = fminf(smin[tid], smin[tid + s]);
            smax[tid] = fmaxf(smax[tid], smax[tid + s]);
        }
        __syncthreads();
    }
    if (tid == 0) {
        float lo = smin[0], hi = smax[0];
        if (hi == lo) { lo -= 1.0f; hi += 1.0f; }   // torch.histc same-value rule
        mn[bc] = lo;
        mx[bc] = hi;
    }
}

// ---------------- Pass 2: fused conv+relu, per-(b,c) 64-bin histogram ----------------
__global__ void conv_hist_kernel(const float* __restrict__ x,
                                 const float* __restrict__ w,
                                 const float* __restrict__ bias,
                                 const float* __restrict__ mn,
                                 const float* __restrict__ mx,
                                 float* __restrict__ feat) {
    const int bc = blockIdx.x;
    const int b  = bc >> 6;
    const int c  = bc & 63;
    const int tid = threadIdx.x;

    __shared__ unsigned int hist[BINS];
    if (tid < BINS) hist[tid] = 0u;
    __syncthreads();

    const float* wc = w + c * 9;
    const float w0 = wc[0], w1 = wc[1], w2 = wc[2];
    const float w3 = wc[3], w4 = wc[4], w5 = wc[5];
    const float w6 = wc[6], w7 = wc[7], w8 = wc[8];
    const float bb = bias[c];
    const float* xb = x + (size_t)b * IW * IW;

    const float lo = mn[bc];
    const float hi = mx[bc];
    const float scale = (float)BINS / (hi - lo);

    for (int p = tid; p < NPIX; p += blockDim.x) {
        const int i = p / OW;
        const int j = p - i * OW;
        const float* r0 = xb + i * IW + j;
        float v = bb;
        v = fmaf(w0, r0[0],       v);
        v = fmaf(w1, r0[1],       v);
        v = fmaf(w2, r0[2],       v);
        v = fmaf(w3, r0[IW + 0],  v);
        v = fmaf(w4, r0[IW + 1],  v);
        v = fmaf(w5, r0[IW + 2],  v);
        v = fmaf(w6, r0[2*IW + 0], v);
        v = fmaf(w7, r0[2*IW + 1], v);
        v = fmaf(w8, r0[2*IW + 2], v);
        v = fmaxf(v, 0.0f);
        int idx = (int)floorf((v - lo) * scale);
        idx = idx < 0 ? 0 : (idx > BINS - 1 ? BINS - 1 : idx);
        atomicAdd(&hist[idx], 1u);
    }
    __syncthreads();
    // feat[b, c*64 + bin] == feat[bc*64 + bin]
    if (tid < BINS) feat[(size_t)bc * BINS + tid] = (float)hist[tid];
}

// ---------------- Head GEMM: out[64,1000] = feat[64,4096] @ head_w.T + head_b ----------------
// V_WMMA_F32_16X16X4_F32, one wave per 16x16 output tile, K-loop over 4096 in steps of 4.
#define GM 64
#define GN 1000
#define GK 4096
#define NT ((GN + 15) / 16)   // 63 n-tiles
#define MT (GM / 16)          // 4 m-tiles
#define NTILES (MT * NT)      // 252 wave-tiles

__global__ void head_gemm_kernel(const float* __restrict__ feat,
                                 const float* __restrict__ hw,
                                 const float* __restrict__ hb,
                                 float* __restrict__ out) {
    const int wave = blockIdx.x * (blockDim.x >> 5) + (threadIdx.x >> 5);
    if (wave >= NTILES) return;            // wave-uniform: EXEC stays all-1s

    const int mt = wave / NT;
    const int nt = wave - mt * NT;
    const int lane = threadIdx.x & 31;
    const int half = lane >> 4;            // 0: K={0,1}, 1: K={2,3}
    const int l15  = lane & 15;

    const int aRow = mt * 16 + l15;                     // M index, always < 64
    const int bCol = nt * 16 + l15;                     // N index, may exceed 999
    const int bColC = bCol < GN ? bCol : GN - 1;        // clamp (garbage cols never stored)

    const float* ap = feat + (size_t)aRow * GK + 2 * half;
    const float* bp = hw   + (size_t)bColC * GK + 2 * half;

    v8f acc = {};
#pragma unroll 4
    for (int k = 0; k < GK; k += 4) {
        v2f a  = *(const v2f*)(ap + k);
        v2f bv = *(const v2f*)(bp + k);
        // D = A(16x4) * B(4x16) + C ; emits v_wmma_f32_16x16x4_f32
        acc = __builtin_amdgcn_wmma_f32_16x16x4_f32(
            /*neg_a=*/false, a, /*neg_b=*/false, bv,
            /*c_mod=*/(short)0, acc, /*reuse_a=*/false, /*reuse_b=*/false);
    }

    // C/D layout: VGPR r -> (M = r + 8*half, N = l15) per lane half
    if (bCol < GN) {
        const float bias = hb[bCol];
#pragma unroll
        for (int r = 0; r < 8; ++r) {
            const int row = mt * 16 + r + 8 * half;
            out[(size_t)row * GN + bCol] = acc[r] + bias;
        }
    }
}

extern "C" void kernel_launch(void* const* d_in, const int* in_sizes, int n_in,
                              void* d_out, int out_size, void* d_ws, size_t ws_size,
                              hipStream_t stream) {
    const float* x      = (const float*)d_in[0];  // [64,1,256,256]
    const float* conv_w = (const float*)d_in[1];  // [64,1,3,3]
    const float* conv_b = (const float*)d_in[2];  // [64]
    const float* head_w = (const float*)d_in[3];  // [1000,4096]
    const float* head_b = (const float*)d_in[4];  // [1000]
    float* out = (float*)d_out;                   // [64,1000]

    // Workspace layout (floats): mn[4096] | mx[4096] | feat[64*4096]
    float* mn   = (float*)d_ws;
    float* mx   = mn + B_N * C_OUT;
    float* feat = mx + B_N * C_OUT;

    conv_minmax_kernel<<<B_N * C_OUT, 256, 0, stream>>>(x, conv_w, conv_b, mn, mx);
    conv_hist_kernel  <<<B_N * C_OUT, 256, 0, stream>>>(x, conv_w, conv_b, mn, mx, feat);

    const int wavesPerBlock = 8;               // 256 threads
    const int blocks = (NTILES + wavesPerBlock - 1) / wavesPerBlock;
    head_gemm_kernel<<<blocks, 256, 0, stream>>>(feat, head_w, head_b, out);
}